// R2D2_7748121002204
// MI455X (gfx1250) — compile-verified
//
#include <hip/hip_runtime.h>
#include <math.h>
#include <stdint.h>
#include <string.h>

#define HH 64
#define WW 64
#define N1 10
#define N2 100
#define BATCHN 256

// Padded layer-2 map: taps reach rows/cols [-31, 94] for every generated
// config (pr,pc <= 31; dilation*(k-1) <= 63), so a 126x126 halo covers all.
#define PAD 31
#define PW (HH + 2 * PAD)   // 126

typedef float v2f __attribute__((ext_vector_type(2)));
typedef float v8f __attribute__((ext_vector_type(8)));

struct Cfg { short kr, kc, dr, dc, pr, pc, Ho, Wo; };
struct CfgTable { Cfg c[N2]; };

// ---------------------------------------------------------------------------
// Kernel A: layer-1 conv (10 filters, 3x3, pad 1) via im2col micro-GEMM on
// V_WMMA_F32_16X16X4_F32.  M = filter (10 of 16 used), K = tap (9 of 12),
// N = 16 consecutive pixels of one image row.  One block per batch element.
// x is staged in a zero-filled 66x66 halo so tap loads are branch-free.
// ---------------------------------------------------------------------------
__global__ __launch_bounds__(256)
void layer1_wmma(const float* __restrict__ x, const float* __restrict__ w1,
                 const float* __restrict__ b1, float* __restrict__ y)
{
    __shared__ float sX[66 * 66];
    const int b   = blockIdx.x;
    const int tid = threadIdx.x;
    const float* xb = x + (size_t)b * HH * WW;
    for (int p = tid; p < 66 * 66; p += 256) {
        const int rr = p / 66 - 1, cc = p % 66 - 1;
        float v = 0.f;
        if ((unsigned)rr < HH && (unsigned)cc < WW) v = xb[rr * WW + cc];
        sX[p] = v;
    }
    __syncthreads();

    const int lane = tid & 31;
    const int wv   = tid >> 5;            // wave id 0..7
    const int half = lane >> 4;           // 0: lanes 0-15, 1: lanes 16-31
    const int n    = lane & 15;           // N index / M index per ISA layouts

    // A matrix (weights), 3 K-chunks of 4 taps.  Per ISA: lane<16 holds K=0,1
    // (VGPR0,VGPR1) for M=lane; lane>=16 holds K=2,3 for M=lane-16.
    v2f a[3];
    const int m = n;
    #pragma unroll
    for (int c = 0; c < 3; ++c) {
        const int k0 = 4 * c + 2 * half;
        const int k1 = k0 + 1;
        float a0 = 0.f, a1 = 0.f;
        if (m < N1 && k0 < 9) a0 = w1[m * 9 + k0];
        if (m < N1 && k1 < 9) a1 = w1[m * 9 + k1];
        a[c][0] = a0; a[c][1] = a1;
    }
    // Per-lane tap offsets into the halo tile for the B matrix (branch-free:
    // invalid taps use a clamped address and select 0 afterwards).
    int  off0[3], off1[3];
    bool ok0[3], ok1[3];
    #pragma unroll
    for (int c = 0; c < 3; ++c) {
        int k0 = 4 * c + 2 * half, k1 = k0 + 1;
        ok0[c] = (k0 < 9); ok1[c] = (k1 < 9);
        if (k0 > 8) k0 = 8;
        if (k1 > 8) k1 = 8;
        off0[c] = (k0 / 3) * 66 + (k0 % 3);
        off1[c] = (k1 / 3) * 66 + (k1 % 3);
    }
    // C (bias) per ISA C/D layout: VGPR v -> M = v + 8*half, N = lane&15.
    v8f cbias;
    #pragma unroll
    for (int v = 0; v < 8; ++v) {
        const int mf = v + 8 * half;
        cbias[v] = (mf < N1) ? b1[mf] : 0.f;
    }

    float* yb = y + (size_t)b * N1 * (HH * WW);

    for (int g = wv; g < (HH * WW) / 16; g += 8) {
        const int r    = g >> 2;                 // image row
        const int col  = ((g & 3) << 4) + n;     // image column for this lane
        const int base = r * 66 + col;           // halo-tile anchor (top-left)
        v2f bm[3];
        #pragma unroll
        for (int c = 0; c < 3; ++c) {
            const float v0 = sX[base + off0[c]];
            const float v1 = sX[base + off1[c]];
            bm[c][0] = ok0[c] ? v0 : 0.f;
            bm[c][1] = ok1[c] ? v1 : 0.f;
        }
        v8f acc = cbias;
        acc = __builtin_amdgcn_wmma_f32_16x16x4_f32(false, a[0], false, bm[0], (short)0, acc, false, false);
        acc = __builtin_amdgcn_wmma_f32_16x16x4_f32(false, a[1], false, bm[1], (short)0, acc, false, false);
        acc = __builtin_amdgcn_wmma_f32_16x16x4_f32(false, a[2], false, bm[2], (short)0, acc, false, false);

        const size_t pix = (size_t)r * WW + col;
        yb[(size_t)(8 * half) * (HH * WW) + pix]     = acc[0];   // filters 0 / 8
        yb[(size_t)(8 * half + 1) * (HH * WW) + pix] = acc[1];   // filters 1 / 9
        if (half == 0) {
            #pragma unroll
            for (int v = 2; v < 8; ++v)
                yb[(size_t)v * (HH * WW) + pix] = acc[v];        // filters 2..7
        }
    }
}

// ---------------------------------------------------------------------------
// PPV sweep over the 100 configs for one (b,i) map resident in a 126x126
// zero-padded LDS tile: the dilated-conv inner loop is fully branch-free and
// division-free (incremental row/col stepping).  Reduction: wave32 shuffle
// butterfly + 8-entry LDS combine, fixed order (deterministic).
// ---------------------------------------------------------------------------
__device__ inline void ppv_sweep(const float* sYP, float* sK, float* sRed8,
                                 const float* __restrict__ w2,
                                 const float* __restrict__ b2,
                                 float* __restrict__ out,
                                 const CfgTable& cfgs, int b, int i, int tid)
{
    const float T = 1e-7f;
    const int lane = tid & 31;
    const int wv   = tid >> 5;
    const float* basep = sYP + PAD * PW + PAD;   // (row 0, col 0) of the map
    for (int j = 0; j < N2; ++j) {
        if (tid < 16) sK[tid] = w2[j * 16 + tid];
        __syncthreads();
        const Cfg cf = cfgs.c[j];
        const int kr = cf.kr, kc = cf.kc, dr = cf.dr, dc = cf.dc;
        const int pr = cf.pr, pc = cf.pc, Ho = cf.Ho, Wo = cf.Wo;
        const int Np = Ho * Wo;
        const float bj = b2[j];
        // stride-256 stepping: 256 = rstep*Wo + cstep, so at most one wrap
        const int rstep = 256 / Wo;
        const int cstep = 256 - rstep * Wo;
        int r = tid / Wo;
        int c = tid - r * Wo;
        float local = 0.f;
        for (int p = tid; p < Np; p += 256) {
            const float* anchor = basep + (r - pr) * PW + (c - pc);
            float z = bj;
            for (int er = 0; er < kr; ++er) {
                const float* rowp = anchor + er * dr * PW;
                for (int ec = 0; ec < kc; ++ec)
                    z = fmaf(sK[er * 4 + ec], rowp[ec * dc], z);
            }
            const float v = fminf(fmaxf(z, -T), T) / T;     // clamp/t as in ref
            local += v * 0.5f + 0.5f;
            c += cstep; r += rstep;
            if (c >= Wo) { c -= Wo; ++r; }
        }
        // wave32 butterfly reduce (no barriers), then combine 8 wave partials
        #pragma unroll
        for (int off = 16; off > 0; off >>= 1)
            local += __shfl_xor(local, off, 32);
        if (lane == 0) sRed8[wv] = local;
        __syncthreads();
        if (tid == 0) {
            float s = sRed8[0];
            #pragma unroll
            for (int w = 1; w < 8; ++w) s += sRed8[w];
            out[(size_t)b * (N1 * N2) + i * N2 + j] = s / (float)Np;
        }
    }
}

// Kernel B: one block per (b, i) map; map staged into padded LDS from y.
__global__ __launch_bounds__(256)
void ppv_from_y(const float* __restrict__ y, const float* __restrict__ w2,
                const float* __restrict__ b2, float* __restrict__ out,
                CfgTable cfgs)
{
    __shared__ float sYP[PW * PW];
    __shared__ float sK[16];
    __shared__ float sRed8[8];
    const int tid = threadIdx.x;
    const int b = blockIdx.x / N1;
    const int i = blockIdx.x % N1;
    const float* ym = y + (size_t)(b * N1 + i) * (HH * WW);
    for (int p = tid; p < PW * PW; p += 256) sYP[p] = 0.f;
    __syncthreads();
    for (int p = tid; p < HH * WW; p += 256)
        sYP[((p >> 6) + PAD) * PW + (p & 63) + PAD] = ym[p];
    __syncthreads();
    ppv_sweep(sYP, sK, sRed8, w2, b2, out, cfgs, b, i, tid);
}

// Fallback: fully fused (scalar layer-1 per map, x read from L2-resident
// global) if the workspace is too small for the staged-y pipeline.
__global__ __launch_bounds__(256)
void ppv_fused(const float* __restrict__ x, const float* __restrict__ w1,
               const float* __restrict__ b1, const float* __restrict__ w2,
               const float* __restrict__ b2, float* __restrict__ out,
               CfgTable cfgs)
{
    __shared__ float sYP[PW * PW];
    __shared__ float sK[16];
    __shared__ float sRed8[8];
    const int tid = threadIdx.x;
    const int b = blockIdx.x / N1;
    const int i = blockIdx.x % N1;
    const float* xb = x + (size_t)b * HH * WW;
    float wk[9];
    #pragma unroll
    for (int t = 0; t < 9; ++t) wk[t] = w1[i * 9 + t];
    const float bi = b1[i];
    for (int p = tid; p < PW * PW; p += 256) sYP[p] = 0.f;
    __syncthreads();
    for (int p = tid; p < HH * WW; p += 256) {
        const int r = p >> 6, c = p & 63;
        float z = bi;
        #pragma unroll
        for (int er = 0; er < 3; ++er) {
            const int rr = r + er - 1;
            #pragma unroll
            for (int ec = 0; ec < 3; ++ec) {
                const int cc = c + ec - 1;
                if ((unsigned)rr < HH && (unsigned)cc < WW)
                    z = fmaf(wk[er * 3 + ec], xb[rr * WW + cc], z);
            }
        }
        sYP[(r + PAD) * PW + c + PAD] = z;
    }
    __syncthreads();
    ppv_sweep(sYP, sK, sRed8, w2, b2, out, cfgs, b, i, tid);
}

// ---------------------------------------------------------------------------
// Host: exact replication of np.random.RandomState(0) config generation.
// ---------------------------------------------------------------------------
namespace {
struct MT19937 {
    uint32_t mt[624]; int mti;
    explicit MT19937(uint32_t s) {
        mt[0] = s;
        for (int i = 1; i < 624; ++i)
            mt[i] = 1812433253u * (mt[i - 1] ^ (mt[i - 1] >> 30)) + (uint32_t)i;
        mti = 624;
    }
    uint32_t u32() {
        if (mti >= 624) {
            for (int i = 0; i < 624; ++i) {
                uint32_t y = (mt[i] & 0x80000000u) | (mt[(i + 1) % 624] & 0x7fffffffu);
                uint32_t v = mt[(i + 397) % 624] ^ (y >> 1);
                if (y & 1u) v ^= 2567483615u;
                mt[i] = v;
            }
            mti = 0;
        }
        uint32_t y = mt[mti++];
        y ^= y >> 11; y ^= (y << 7) & 2636928640u;
        y ^= (y << 15) & 4022730752u; y ^= y >> 18;
        return y;
    }
    uint64_t u64() { uint64_t hi = u32(); uint64_t lo = u32(); return (hi << 32) | lo; }
    double d53() {
        uint32_t a = u32() >> 5, b = u32() >> 6;
        return (a * 67108864.0 + b) / 9007199254740992.0;
    }
    uint64_t bounded(uint64_t rng) {   // inclusive max, masked rejection (numpy legacy int64)
        if (!rng) return 0;
        uint64_t mask = rng;
        mask |= mask >> 1; mask |= mask >> 2; mask |= mask >> 4;
        mask |= mask >> 8; mask |= mask >> 16; mask |= mask >> 32;
        uint64_t v;
        do { v = u64() & mask; } while (v > rng);
        return v;
    }
};

void make_cfgs(CfgTable& t) {
    MT19937 rng(0);
    for (int j = 0; j < N2; ++j) {
        const int kr = 2 + (int)rng.bounded(2);   // choice((2,3,4)) -> randint(0,3)
        const int kc = 2 + (int)rng.bounded(2);
        const double ur = rng.d53() * log2((double)(HH - 1) / (double)(kr - 1));
        const int dr = (int)pow(2.0, ur);
        const double uc = rng.d53() * log2((double)(WW - 1) / (double)(kc - 1));
        const int dc = (int)pow(2.0, uc);
        int pr = 0, pc = 0;
        if (rng.bounded(1) == 1) {                // randint(2)
            pr = (kr - 1) * dr / 2;
            pc = (kc - 1) * dc / 2;
        }
        const int Ho = HH + 2 * pr - dr * (kr - 1);
        const int Wo = WW + 2 * pc - dc * (kc - 1);
        t.c[j] = { (short)kr, (short)kc, (short)dr, (short)dc,
                   (short)pr, (short)pc, (short)Ho, (short)Wo };
    }
}
} // namespace

extern "C" void kernel_launch(void* const* d_in, const int* in_sizes, int n_in,
                              void* d_out, int out_size, void* d_ws, size_t ws_size,
                              hipStream_t stream) {
    (void)in_sizes; (void)n_in; (void)out_size;
    const float* x  = (const float*)d_in[0];
    const float* w1 = (const float*)d_in[1];
    const float* b1 = (const float*)d_in[2];
    const float* w2 = (const float*)d_in[3];
    const float* b2 = (const float*)d_in[4];
    float* out = (float*)d_out;

    CfgTable tbl;
    make_cfgs(tbl);

    const size_t needed = (size_t)BATCHN * N1 * HH * WW * sizeof(float);
    if (ws_size >= needed) {
        float* y = (float*)d_ws;
        layer1_wmma<<<BATCHN, 256, 0, stream>>>(x, w1, b1, y);
        ppv_from_y<<<BATCHN * N1, 256, 0, stream>>>(y, w2, b2, out, tbl);
    } else {
        ppv_fused<<<BATCHN * N1, 256, 0, stream>>>(x, w1, b1, w2, b2, out, tbl);
    }
}